// HashEmbedding_44976897523736
// MI455X (gfx1250) — compile-verified
//
#include <hip/hip_runtime.h>
#include <hip/hip_bf16.h>
#include <stdint.h>

// Problem geometry (fixed by the reference).
#define TOKENS        (16 * 4096)          // 65536
#define NUM_HASHES    4
#define EMB_DIM       256
#define KVAL          512
#define WTAB_FLOATS   (NUM_HASHES * (KVAL + 1))   // 2052
#define EMB_ROWS      257
#define S_EMB_FLOATS  (EMB_ROWS * EMB_DIM)        // 65792
#define LDS_BYTES     ((S_EMB_FLOATS + WTAB_FLOATS) * 4)  // 271,376 B < 320 KiB

// Launch geometry.
#define BLOCKS        256
#define THREADS       512                          // 16 wave32 waves
#define WAVES         (THREADS / 32)
#define TOK_PER_BLOCK (TOKENS / BLOCKS)            // 256
#define TOK_PER_WAVE  (TOK_PER_BLOCK / WAVES)      // 16

__device__ __forceinline__ float4 wsum4(float w0, float4 a,
                                        float w1, float4 b,
                                        float w2, float4 c,
                                        float w3, float4 d) {
  float4 r;
  r.x = fmaf(w3, d.x, fmaf(w2, c.x, fmaf(w1, b.x, w0 * a.x)));
  r.y = fmaf(w3, d.y, fmaf(w2, c.y, fmaf(w1, b.y, w0 * a.y)));
  r.z = fmaf(w3, d.z, fmaf(w2, c.z, fmaf(w1, b.z, w0 * a.z)));
  r.w = fmaf(w3, d.w, fmaf(w2, c.w, fmaf(w1, b.w, w0 * a.w)));
  return r;
}

// Issue one 16-byte async global->LDS copy (CDNA5 VGLOBAL async op, ASYNCcnt).
__device__ __forceinline__ void async_copy_b128(uint32_t lds_byte_addr,
                                                const void* gaddr) {
  asm volatile("global_load_async_to_lds_b128 %0, %1, off"
               :
               : "v"(lds_byte_addr), "v"(gaddr)
               : "memory");
}

__global__ __launch_bounds__(THREADS)
void hash_embedding_kernel(const int* __restrict__ x,          // [TOKENS,4] int32
                           const float* __restrict__ wtab,     // [2052]
                           const float* __restrict__ etab,     // [257,256]
                           float* __restrict__ out)            // [TOKENS,256]
{
  extern __shared__ float smem[];
  float* s_emb = smem;                       // 65792 floats
  float* s_w   = smem + S_EMB_FLOATS;        // 2052 floats

  // LDS byte offset of dynamic shared base (should be 0; computed robustly).
  const uint32_t lds0 =
      (uint32_t)(uintptr_t)(__attribute__((address_space(3))) float*)smem;

  const int tid = threadIdx.x;

  // ---- Stage entire embedding table + weight table into LDS via async DMA ----
  #pragma unroll 1
  for (int c = tid; c < S_EMB_FLOATS / 4; c += THREADS) {
    async_copy_b128(lds0 + 16u * (uint32_t)c, etab + 4 * c);
  }
  #pragma unroll 1
  for (int c = tid; c < WTAB_FLOATS / 4; c += THREADS) {
    async_copy_b128(lds0 + (uint32_t)(S_EMB_FLOATS * 4) + 16u * (uint32_t)c,
                    wtab + 4 * c);
  }

  // Drain this wave's async copies, then workgroup barrier so all waves see LDS.
#if __has_builtin(__builtin_amdgcn_s_wait_asynccnt)
  __builtin_amdgcn_s_wait_asynccnt(0);
#else
  asm volatile("s_wait_asynccnt 0" ::: "memory");
#endif
  __syncthreads();

  // ---- Main loop: one wave per token; lane l covers dims [4l,4l+4) and [128+4l,...)
  const int lane = tid & 31;
  // Wave id is uniform across the wave; make that explicit so the x-descriptor
  // load and per-hash base addresses become scalar (SMEM/SALU) work.
  const int wave = __builtin_amdgcn_readfirstlane(tid >> 5);
  const int tok_base = blockIdx.x * TOK_PER_BLOCK + wave * TOK_PER_WAVE;

  #pragma unroll 4
  for (int t = 0; t < TOK_PER_WAVE; ++t) {
    const int token = tok_base + t;

    // 4 hash ids for this token (wave-uniform).
    const int4 xi = *reinterpret_cast<const int4*>(x + 4 * (size_t)token);

    // Per-sample weights: weight_idx = x[h] + h*(K+1)
    const float w0 = s_w[xi.x];
    const float w1 = s_w[xi.y + 1 * (KVAL + 1)];
    const float w2 = s_w[xi.z + 2 * (KVAL + 1)];
    const float w3 = s_w[xi.w + 3 * (KVAL + 1)];

    // Embedding rows: idx = x[h] >> 1, row stride 256 floats.
    const float4* e0 = (const float4*)(s_emb + (((unsigned)xi.x >> 1) << 8));
    const float4* e1 = (const float4*)(s_emb + (((unsigned)xi.y >> 1) << 8));
    const float4* e2 = (const float4*)(s_emb + (((unsigned)xi.z >> 1) << 8));
    const float4* e3 = (const float4*)(s_emb + (((unsigned)xi.w >> 1) << 8));

    const float4 lo = wsum4(w0, e0[lane],      w1, e1[lane],
                            w2, e2[lane],      w3, e3[lane]);
    const float4 hi = wsum4(w0, e0[lane + 32], w1, e1[lane + 32],
                            w2, e2[lane + 32], w3, e3[lane + 32]);

    // Coalesced: each b128 store from the wave covers a contiguous 512B span.
    float4* o = (float4*)(out + (size_t)token * EMB_DIM);
    o[lane]      = lo;
    o[lane + 32] = hi;
  }
}

extern "C" void kernel_launch(void* const* d_in, const int* in_sizes, int n_in,
                              void* d_out, int out_size, void* d_ws, size_t ws_size,
                              hipStream_t stream) {
  (void)in_sizes; (void)n_in; (void)out_size; (void)d_ws; (void)ws_size;

  const int*   x    = (const int*)d_in[0];    // [16,4096,4] int
  const float* wtab = (const float*)d_in[1];  // [2052,1] f32
  const float* etab = (const float*)d_in[2];  // [257,256] f32
  float*       out  = (float*)d_out;          // [16,4096,256] f32

  // Allow >default dynamic LDS (271,376 B; gfx1250 WGP supports 320 KiB).
  (void)hipFuncSetAttribute((const void*)hash_embedding_kernel,
                            hipFuncAttributeMaxDynamicSharedMemorySize,
                            LDS_BYTES);

  hash_embedding_kernel<<<dim3(BLOCKS), dim3(THREADS), LDS_BYTES, stream>>>(
      x, wtab, etab, out);
}